// RPNBaseModel_58076547777054
// MI455X (gfx1250) — compile-verified
//
#include <hip/hip_runtime.h>
#include <hip/hip_bf16.h>
#include <stdint.h>

// ---------------------------------------------------------------------------
// RPN proposal layer for MI455X (gfx1250, wave32).
//   stage 1: radix-threshold top-k candidate selection (per-image block)
//   stage 2: bitonic sort of 8192 u64 keys in LDS (keys staged with
//            global_load_async_to_lds_b128) + box decode/clip
//   stage 3: NMS suppression bit-matrix; one wave32 per row, words formed
//            with __ballot; 32KB box tiles DMA'd into LDS by the CDNA5
//            Tensor Data Mover (tensor_load_to_lds + s_wait_tensorcnt)
//   stage 4: serial greedy NMS reduce, one wave32 per image, removed-mask
//            resident in 6 VGPRs/lane, rows prefetched (global_prefetch_b8)
// 19MB workspace -> fully resident in MI455X's 192MB L2.
// ---------------------------------------------------------------------------

#define KSEL   6000     // pre_nms_limit (matches reference setup)
#define PCNT   1000     // proposal_count (matches reference setup)
#define NMSW   192      // padded u32 words per suppression row (covers 6144)
#define NMSTHR 0.7f
#define IMGDIM 1024.0f

typedef unsigned int u32x4 __attribute__((ext_vector_type(4)));
typedef int          i32x8 __attribute__((ext_vector_type(8)));
typedef int          i32x4 __attribute__((ext_vector_type(4)));

// ---- CDNA5 async global->LDS helpers (gfx1250-specific) -------------------
__device__ __forceinline__ void async_copy_b128(void* lds_dst, const void* gsrc) {
  // low 32 bits of a flat shared pointer are the LDS byte address (ISA 10.2)
  unsigned lds_off = (unsigned)(uintptr_t)lds_dst;
  unsigned long long ga = (unsigned long long)(uintptr_t)gsrc;
  asm volatile("global_load_async_to_lds_b128 %0, %1, off"
               :: "v"(lds_off), "v"(ga) : "memory");
}
__device__ __forceinline__ void wait_async0() {
  asm volatile("s_wait_asynccnt 0" ::: "memory");
}

// ---- Tensor Data Mover: 1-D tile of n_elem 4-byte elements ----------------
// D# layout per CDNA5 ISA ch.8 (group0: count/lds_addr/global_addr/type,
// group1: data_size + dims/strides). Issued by one wave; EXEC is ignored by
// TDM, completion tracked with TENSORcnt. 6-arg builtin (clang-23 form).
__device__ __forceinline__ void tdm_load_1d_f32(void* lds_dst, const void* gsrc,
                                                unsigned n_elem) {
#if __has_builtin(__builtin_amdgcn_tensor_load_to_lds)
  unsigned long long ga = (unsigned long long)(uintptr_t)gsrc;
  u32x4 g0;
  g0[0] = 1u;                                        // count=1, no gather
  g0[1] = (unsigned)(uintptr_t)lds_dst;              // lds_addr
  g0[2] = (unsigned)(ga & 0xffffffffull);            // global_addr[31:0]
  g0[3] = (unsigned)((ga >> 32) & 0x01ffffffull)     // global_addr[56:32]
        | (2u << 30);                                // type=2 ("image")
  i32x8 g1;
  g1[0] = 0x00020000;                                // data_size=2 (4 bytes)
  g1[1] = (int)((n_elem & 0xffffu) << 16);           // tensor_dim0[15:0]
  g1[2] = (int)((n_elem >> 16) | (1u << 16));        // dim0[31:16] | dim1=1
  g1[3] = (int)((n_elem & 0xffffu) << 16);           // tile_dim0 = n_elem
  g1[4] = 1;                                         // tile_dim1=1, tile_dim2=0
  g1[5] = (int)n_elem;                               // dim0_stride[31:0]
  g1[6] = (int)((n_elem & 0xffffu) << 16);           // stride0 hi | stride1 lo
  g1[7] = (int)(n_elem >> 16);                       // stride1[47:16]
  i32x4 g2 = {0, 0, 0, 0};
  i32x4 g3 = {0, 0, 0, 0};
  i32x8 g4 = {0, 0, 0, 0, 0, 0, 0, 0};
  __builtin_amdgcn_tensor_load_to_lds(g0, g1, g2, g3, g4, 0);
  __builtin_amdgcn_s_wait_tensorcnt(0);
#else
  // fallback: 32 lanes of this wave async-copy the tile in b128 chunks
  const uint4* src = (const uint4*)gsrc;
  uint4* dst = (uint4*)lds_dst;
  unsigned nchunk = n_elem / 4;
  for (unsigned c = threadIdx.x & 31; c < nchunk; c += 32)
    async_copy_b128(&dst[c], &src[c]);
  wait_async0();
#endif
}

// ---- float <-> order-preserving unsigned ----------------------------------
__device__ __forceinline__ unsigned flipf(float f) {
  unsigned u = __float_as_uint(f);
  return (u & 0x80000000u) ? ~u : (u | 0x80000000u);
}
__device__ __forceinline__ float unflipf(unsigned u) {
  unsigned v = (u & 0x80000000u) ? (u & 0x7fffffffu) : ~u;
  return __uint_as_float(v);
}

// ---------------------------------------------------------------------------
__global__ void zero_out_kernel(float* out, int n) {
  int i = blockIdx.x * 256 + threadIdx.x;
  if (i < n) out[i] = 0.0f;
}

// ---------------------------------------------------------------------------
// Stage 1: per-image top-k candidate selection via 24-bit radix threshold.
// Candidate set (everything >= 24-bit threshold) is value-determined, hence
// deterministic; ordering is fixed by the sort in stage 2.
// ---------------------------------------------------------------------------
__global__ void __launch_bounds__(1024)
topk_select_kernel(const float* __restrict__ probs,
                   unsigned long long* __restrict__ keys, int A) {
  const int b = blockIdx.x, tid = threadIdx.x, NT = 1024;
  __shared__ unsigned hist[4096];
  __shared__ unsigned sT1, sNeed, sThr24, sCnt;

  const float* p = probs + (size_t)b * A * 2;

  // level 1: histogram of top 12 bits
  for (int s = tid; s < 4096; s += NT) hist[s] = 0;
  __syncthreads();
  for (int i = tid; i < A; i += NT) {
    unsigned u = flipf(p[2 * i + 1]);
    atomicAdd(&hist[u >> 20], 1u);
  }
  __syncthreads();
  if (tid == 0) {
    unsigned cum = 0; int t1 = 0; unsigned need = KSEL;
    for (int s = 4095; s >= 0; --s) {
      unsigned c = hist[s];
      if (cum + c >= KSEL) { t1 = s; need = KSEL - cum; break; }
      cum += c;
    }
    sT1 = (unsigned)t1; sNeed = need;
  }
  __syncthreads();
  const unsigned T1 = sT1, need = sNeed;

  // level 2: histogram of next 12 bits within bin T1
  for (int s = tid; s < 4096; s += NT) hist[s] = 0;
  __syncthreads();
  for (int i = tid; i < A; i += NT) {
    unsigned u = flipf(p[2 * i + 1]);
    if ((u >> 20) == T1) atomicAdd(&hist[(u >> 8) & 0xFFFu], 1u);
  }
  __syncthreads();
  if (tid == 0) {
    unsigned cum = 0; int t2 = 0;
    for (int s = 4095; s >= 0; --s) {
      unsigned c = hist[s];
      if (cum + c >= need) { t2 = s; break; }
      cum += c;
    }
    sThr24 = (T1 << 12) | (unsigned)t2;
    sCnt = 0;
  }
  __syncthreads();
  const unsigned thr24 = sThr24;

  // pass 3: compact every candidate with 24-bit prefix >= threshold
  unsigned long long* kb = keys + (size_t)b * 8192;
  for (int i = tid; i < A; i += NT) {
    unsigned u = flipf(p[2 * i + 1]);
    if ((u >> 8) >= thr24) {
      unsigned pos = atomicAdd(&sCnt, 1u);
      if (pos < 8192)   // descending sort => ~i makes ties prefer smaller idx
        kb[pos] = ((unsigned long long)u << 32) | (unsigned)(~i);
    }
  }
  __syncthreads();
  unsigned n = sCnt; if (n > 8192) n = 8192;
  for (unsigned s = n + tid; s < 8192; s += NT) kb[s] = 0ull;
}

// ---------------------------------------------------------------------------
// Stage 2: bitonic sort 8192 u64 keys (64KB LDS, fine inside 320KB/WGP),
// then decode deltas -> clipped boxes for the first KSEL entries. Keys are
// staged into LDS via the gfx1250 async global->LDS path.
// ---------------------------------------------------------------------------
__global__ void __launch_bounds__(1024)
sort_decode_kernel(const unsigned long long* __restrict__ keys,
                   const float* __restrict__ deltas,
                   const float* __restrict__ anchors,
                   float* __restrict__ boxesOut,
                   float* __restrict__ scoresOut, int A) {
  __shared__ __align__(16) unsigned long long sk[8192];
  const int b = blockIdx.x, tid = threadIdx.x, NT = 1024;

  // async-stage 64KB of keys: 4096 x b128, 4 per thread
  {
    const uint4* src = (const uint4*)(keys + (size_t)b * 8192);
    uint4* dst = (uint4*)sk;
    for (int e = 0; e < 4; ++e) {
      int c = tid + e * NT;
      async_copy_b128(&dst[c], &src[c]);
    }
    wait_async0();
  }

  for (unsigned kk = 2; kk <= 8192u; kk <<= 1) {
    for (unsigned j = kk >> 1; j > 0; j >>= 1) {
      __syncthreads();
      for (int e = 0; e < 8; ++e) {
        unsigned i = (unsigned)(tid + e * NT);
        unsigned x = i ^ j;
        if (x > i) {
          unsigned long long a = sk[i], c = sk[x];
          bool descend = ((i & kk) == 0);
          if (descend ? (a < c) : (a > c)) { sk[i] = c; sk[x] = a; }
        }
      }
    }
  }
  __syncthreads();

  for (int e = 0; e < 8; ++e) {
    int i = tid + e * NT;
    if (i >= KSEL) continue;
    unsigned long long key = sk[i];
    unsigned u   = (unsigned)(key >> 32);
    unsigned idx = ~(unsigned)key;
    float score = unflipf(u);
    float4 a4 = ((const float4*)anchors)[idx];
    float4 d4 = ((const float4*)deltas)[(size_t)b * A + idx];
    float dy = d4.x * 0.1f, dx = d4.y * 0.1f;
    float dh = d4.z * 0.2f, dw = d4.w * 0.2f;
    float h = a4.z - a4.x, w = a4.w - a4.y;
    float cy = a4.x + 0.5f * h + dy * h;
    float cx = a4.y + 0.5f * w + dx * w;
    h *= expf(dh);
    w *= expf(dw);
    float y1 = cy - 0.5f * h, x1 = cx - 0.5f * w;
    float y2 = y1 + h, x2 = x1 + w;
    y1 = fminf(fmaxf(y1, 0.0f), IMGDIM);
    x1 = fminf(fmaxf(x1, 0.0f), IMGDIM);
    y2 = fminf(fmaxf(y2, 0.0f), IMGDIM);
    x2 = fminf(fmaxf(x2, 0.0f), IMGDIM);
    ((float4*)boxesOut)[(size_t)b * KSEL + i] = make_float4(y1, x1, y2, x2);
    scoresOut[(size_t)b * KSEL + i] = score;
  }
}

// ---------------------------------------------------------------------------
// Stage 3: suppression bit-matrix. One wave32 per row i; bits mark j>i with
// IoU>thr (IoU symmetric => equivalent to the reference's j<i test). Lanes
// evaluate 32 consecutive j's and __ballot forms the word in one op.
// Column tiles (2048 boxes, 32KB) are DMA'd into LDS by the TDM.
// ---------------------------------------------------------------------------
__global__ void __launch_bounds__(1024)
nms_mask_kernel(const float* __restrict__ boxes, unsigned* __restrict__ supp) {
  __shared__ __align__(16) float4 sb[2048];
  const int b = blockIdx.y, tid = threadIdx.x;
  const int wave = tid >> 5, lane = tid & 31;
  const int i = blockIdx.x * 32 + wave;          // one row per wave32
  const float4* bx = (const float4*)boxes + (size_t)b * KSEL;

  float4 bi = make_float4(0.f, 0.f, 0.f, 0.f);
  float ai = 0.0f;
  if (i < KSEL) { bi = bx[i]; ai = (bi.z - bi.x) * (bi.w - bi.y); }
  unsigned* row = supp + ((size_t)b * KSEL + (size_t)(i < KSEL ? i : 0)) * NMSW;

  for (int t = 0; t < 3; ++t) {
    __syncthreads();                              // previous tile consumed
    if (tid < 32) {                               // wave 0 drives the TDM
      unsigned n_elem = (unsigned)(4 * min(KSEL - t * 2048, 2048));
      tdm_load_1d_f32(&sb[0], &bx[t * 2048], n_elem);
    }
    __syncthreads();                              // tile visible to all waves
    if (i < KSEL) {
      for (int jw = 0; jw < 64; ++jw) {
        const int base = t * 2048 + jw * 32;
        unsigned word = 0;
        if (base + 31 > i) {                      // word can hold some j > i
          const int j = base + lane;
          bool pred = false;
          if (j > i && j < KSEL) {
            float4 bj = sb[jw * 32 + lane];
            float yy1 = fmaxf(bi.x, bj.x), xx1 = fmaxf(bi.y, bj.y);
            float yy2 = fminf(bi.z, bj.z), xx2 = fminf(bi.w, bj.w);
            float inter = fmaxf(yy2 - yy1, 0.0f) * fmaxf(xx2 - xx1, 0.0f);
            float aj = (bj.z - bj.x) * (bj.w - bj.y);
            float uni = fmaxf(ai + aj - inter, 1e-10f);
            pred = inter > NMSTHR * uni;
          }
          word = (unsigned)__ballot(pred);
        }
        if (lane == 0) row[t * 64 + jw] = word;
      }
    }
  }
}

// ---------------------------------------------------------------------------
// Stage 4: serial greedy reduce. One wave32 per image; removed-mask (192
// words) lives in 6 VGPRs per lane; decision bit broadcast via shuffle.
// ---------------------------------------------------------------------------
__global__ void __launch_bounds__(32)
nms_reduce_kernel(const float* __restrict__ boxes,
                  const float* __restrict__ scores,
                  const unsigned* __restrict__ supp,
                  float* __restrict__ out, int batch) {
  const int b = blockIdx.x, lane = threadIdx.x;
  unsigned rem0 = 0, rem1 = 0, rem2 = 0, rem3 = 0, rem4 = 0, rem5 = 0;

  float* outBoxes  = out;                                    // [batch,PCNT,4]
  float* outScores = out + (size_t)batch * PCNT * 4;         // [batch,PCNT]
  int*   outCnt    = (int*)(outScores + (size_t)batch * PCNT);

  const unsigned* sp = supp + (size_t)b * KSEL * NMSW;
  int nk = 0;
  for (int i = 0; i < KSEL; ++i) {
    const unsigned* rp = sp + (size_t)i * NMSW;
    if (i + 1 < KSEL) __builtin_prefetch(rp + NMSW, 0, 0);   // global_prefetch
    unsigned r0 = rp[lane],        r1 = rp[32 + lane],  r2 = rp[64 + lane];
    unsigned r3 = rp[96 + lane],   r4 = rp[128 + lane], r5 = rp[160 + lane];

    const int wi = i >> 5;              // owning word index (uniform)
    unsigned sel;
    switch (wi >> 5) {
      case 0:  sel = rem0; break;
      case 1:  sel = rem1; break;
      case 2:  sel = rem2; break;
      case 3:  sel = rem3; break;
      case 4:  sel = rem4; break;
      default: sel = rem5; break;
    }
    unsigned w = __shfl(sel, wi & 31, 32);
    bool suppressed = (w >> (i & 31)) & 1u;
    if (!suppressed) {
      rem0 |= r0; rem1 |= r1; rem2 |= r2; rem3 |= r3; rem4 |= r4; rem5 |= r5;
      if (nk < PCNT && lane == 0) {
        float4 bb = ((const float4*)boxes)[(size_t)b * KSEL + i];
        const float inv = 1.0f / IMGDIM;
        ((float4*)outBoxes)[(size_t)b * PCNT + nk] =
            make_float4(bb.x * inv, bb.y * inv, bb.z * inv, bb.w * inv);
        outScores[(size_t)b * PCNT + nk] = scores[(size_t)b * KSEL + i];
      }
      nk++;
    }
  }
  if (lane == 0) outCnt[b] = (nk < PCNT) ? nk : PCNT;
}

// ---------------------------------------------------------------------------
extern "C" void kernel_launch(void* const* d_in, const int* in_sizes, int n_in,
                              void* d_out, int out_size, void* d_ws, size_t ws_size,
                              hipStream_t stream) {
  const float* probs   = (const float*)d_in[0];   // (batch, A, 2)
  const float* deltas  = (const float*)d_in[1];   // (batch, A, 4)
  const float* anchors = (const float*)d_in[2];   // (A, 4)

  const int A     = in_sizes[2] / 4;
  const int batch = in_sizes[0] / (A * 2);

  // workspace carve-up (all 16B aligned); ~19.2MB for batch=4 -> L2 resident
  char* ws = (char*)d_ws;
  size_t off = 0;
  unsigned long long* keys = (unsigned long long*)(ws + off);
  off += (size_t)batch * 8192 * sizeof(unsigned long long);
  float* scoresS = (float*)(ws + off);
  off += (size_t)batch * KSEL * sizeof(float);
  off = (off + 15) & ~(size_t)15;
  float* boxesS = (float*)(ws + off);
  off += (size_t)batch * KSEL * 4 * sizeof(float);
  unsigned* suppM = (unsigned*)(ws + off);

  float* out = (float*)d_out;

  zero_out_kernel<<<(out_size + 255) / 256, 256, 0, stream>>>(out, out_size);
  topk_select_kernel<<<batch, 1024, 0, stream>>>(probs, keys, A);
  sort_decode_kernel<<<batch, 1024, 0, stream>>>(keys, deltas, anchors,
                                                 boxesS, scoresS, A);
  dim3 mg((KSEL + 31) / 32, batch);
  nms_mask_kernel<<<mg, 1024, 0, stream>>>(boxesS, suppM);
  nms_reduce_kernel<<<batch, 32, 0, stream>>>(boxesS, scoresS, suppM, out, batch);
}